// Pooling_67104569032949
// MI455X (gfx1250) — compile-verified
//
#include <hip/hip_runtime.h>
#include <stdint.h>

#ifndef __has_builtin
#define __has_builtin(x) 0
#endif

#define HH 112
#define WW 112
#define CC 96
#define NN 16
#define ROWS 16
#define HALO 3
#define TR (ROWS + 2 * HALO)   /* 22 rows per stripe incl. halo */
#define PLANE (TR * WW)        /* 2464 LDS elements per plane    */
#define NSTRIPE (HH / ROWS)    /* 7 */

typedef __attribute__((ext_vector_type(4))) unsigned int uint32x4;
typedef __attribute__((ext_vector_type(8))) int          int32x8;
typedef __attribute__((ext_vector_type(4))) int          int32x4;

__global__ __launch_bounds__(256) void pool_diff_kernel(const float* __restrict__ x,
                                                        float* __restrict__ out) {
  __shared__ float sin_[PLANE];
  __shared__ float h3s[PLANE];
  __shared__ float h5s[PLANE];
  __shared__ float h7s[PLANE];

  const int tid    = threadIdx.x;
  const int bid    = blockIdx.x;
  const int stripe = bid % NSTRIPE;
  const int plane  = bid / NSTRIPE;        // n*CC + c
  const int r0     = stripe * ROWS;

  const int row_lo    = r0 - HALO;
  const int row_start = row_lo < 0 ? 0 : row_lo;
  int row_end         = r0 + ROWS + HALO;
  if (row_end > HH) row_end = HH;
  const int nrows = row_end - row_start;
  const int zlo   = row_start - row_lo;          // top halo rows to zero
  const int zhi   = TR - (zlo + nrows);          // bottom halo rows to zero

  // Zero out-of-image halo rows (matches count_include_pad=False: padded
  // elements contribute 0 to the sum; counts are computed analytically).
  for (int i = tid; i < zlo * WW; i += 256) sin_[i] = 0.0f;
  for (int i = tid; i < zhi * WW; i += 256) sin_[(zlo + nrows) * WW + i] = 0.0f;

  const float* gsrc = x + ((size_t)plane * HH + row_start) * WW;

#if __has_builtin(__builtin_amdgcn_tensor_load_to_lds)
  // --- Tensor Data Mover: DMA the valid rows of the stripe into LDS ---
  if (tid < 32) {  // one TDM issue per workgroup (wave 0)
    const uint64_t gaddr = (uint64_t)(uintptr_t)gsrc;
    const uint32_t ldsa  = (uint32_t)(uintptr_t)(&sin_[zlo * WW]);

    uint32x4 g0;
    g0.x = 1u;                                   // count=1, user descriptor
    g0.y = ldsa;                                 // lds_addr (bytes)
    g0.z = (uint32_t)gaddr;                      // global_addr[31:0]
    g0.w = (uint32_t)((gaddr >> 32) & 0x01FFFFFFu) | (2u << 30);  // addr[56:32] | type=2

    int32x8 g1;
    g1[0] = (int)(2u << 16);                     // data_size = 4 bytes
    g1[1] = (int)((uint32_t)WW << 16);           // tensor_dim0[15:0] = 112
    g1[2] = (int)((uint32_t)nrows << 16);        // tensor_dim1[15:0] = nrows
    g1[3] = (int)((uint32_t)WW << 16);           // tile_dim0 = 112
    g1[4] = nrows;                               // tile_dim1 = nrows (tile_dim2=0)
    g1[5] = WW;                                  // tensor_dim0_stride = 112
    g1[6] = 0;
    g1[7] = 0;

    int32x4 gz4; gz4[0] = 0; gz4[1] = 0; gz4[2] = 0; gz4[3] = 0;  // groups 2/3: NULL (2D)
    int32x8 gz8;
    gz8[0] = 0; gz8[1] = 0; gz8[2] = 0; gz8[3] = 0;
    gz8[4] = 0; gz8[5] = 0; gz8[6] = 0; gz8[7] = 0;

    __builtin_amdgcn_tensor_load_to_lds(g0, g1, gz4, gz4, gz8, 0);
    __builtin_amdgcn_s_wait_tensorcnt(0);
  }
#else
  for (int i = tid; i < nrows * WW; i += 256) sin_[zlo * WW + i] = gsrc[i];
#endif
  __syncthreads();

  // --- Pass A: horizontal windowed sums (shared across k = 3/5/7) ---
  for (int i = tid; i < PLANE; i += 256) {
    const int r = i / WW;
    const int w = i - r * WW;
    const float* row = &sin_[r * WW];
    const float c0 = row[w];
    const float l1 = (w >= 1)      ? row[w - 1] : 0.0f;
    const float r1 = (w <= WW - 2) ? row[w + 1] : 0.0f;
    const float l2 = (w >= 2)      ? row[w - 2] : 0.0f;
    const float r2 = (w <= WW - 3) ? row[w + 2] : 0.0f;
    const float l3 = (w >= 3)      ? row[w - 3] : 0.0f;
    const float r3 = (w <= WW - 4) ? row[w + 3] : 0.0f;
    const float s3 = l1 + c0 + r1;
    const float s5 = s3 + l2 + r2;
    const float s7 = s5 + l3 + r3;
    h3s[i] = s3;
    h5s[i] = s5;
    h7s[i] = s7;
  }
  __syncthreads();

  // --- Pass B: vertical sums, normalize by valid count, subtract center ---
  const int n = plane / CC;
  const int c = plane - n * CC;
  const size_t step = (size_t)CC * HH * WW;      // one output channel-block

  for (int i = tid; i < ROWS * WW; i += 256) {
    const int ro = i / WW;
    const int w  = i - ro * WW;
    const int h  = r0 + ro;                      // HH % ROWS == 0 -> always valid
    const int lr = ro + HALO;
    const int b  = lr * WW + w;

    const float v3 = h3s[b - WW] + h3s[b] + h3s[b + WW];
    const float v5 = h5s[b - 2 * WW] + h5s[b - WW] + h5s[b] + h5s[b + WW] + h5s[b + 2 * WW];
    const float v7 = h7s[b - 3 * WW] + h7s[b - 2 * WW] + h7s[b - WW] + h7s[b] +
                     h7s[b + WW] + h7s[b + 2 * WW] + h7s[b + 3 * WW];
    const float xc = sin_[b];

    const int wl1 = (w - 1 < 0) ? 0 : w - 1, wr1 = (w + 1 > WW - 1) ? WW - 1 : w + 1;
    const int wl2 = (w - 2 < 0) ? 0 : w - 2, wr2 = (w + 2 > WW - 1) ? WW - 1 : w + 2;
    const int wl3 = (w - 3 < 0) ? 0 : w - 3, wr3 = (w + 3 > WW - 1) ? WW - 1 : w + 3;
    const int hl1 = (h - 1 < 0) ? 0 : h - 1, hr1 = (h + 1 > HH - 1) ? HH - 1 : h + 1;
    const int hl2 = (h - 2 < 0) ? 0 : h - 2, hr2 = (h + 2 > HH - 1) ? HH - 1 : h + 2;
    const int hl3 = (h - 3 < 0) ? 0 : h - 3, hr3 = (h + 3 > HH - 1) ? HH - 1 : h + 3;
    const int cnt3 = (wr1 - wl1 + 1) * (hr1 - hl1 + 1);
    const int cnt5 = (wr2 - wl2 + 1) * (hr2 - hl2 + 1);
    const int cnt7 = (wr3 - wl3 + 1) * (hr3 - hl3 + 1);

    const size_t obase = ((((size_t)n * 4) * CC + c) * HH + h) * WW + w;

    // Streaming outputs (308 MB, written once) -> non-temporal stores.
    __builtin_nontemporal_store(0.0f, &out[obase]);
    __builtin_nontemporal_store(v3 / (float)cnt3 - xc, &out[obase + step]);
    __builtin_nontemporal_store(v5 / (float)cnt5 - xc, &out[obase + 2 * step]);
    __builtin_nontemporal_store(v7 / (float)cnt7 - xc, &out[obase + 3 * step]);
  }
}

extern "C" void kernel_launch(void* const* d_in, const int* in_sizes, int n_in,
                              void* d_out, int out_size, void* d_ws, size_t ws_size,
                              hipStream_t stream) {
  (void)in_sizes; (void)n_in; (void)out_size; (void)d_ws; (void)ws_size;
  const float* x = (const float*)d_in[0];
  float* out = (float*)d_out;
  const int grid = NN * CC * NSTRIPE;   // 16 * 96 * 7 = 10752 workgroups
  pool_diff_kernel<<<grid, 256, 0, stream>>>(x, out);
}